// MultiHeadAttention_7816840479165
// MI455X (gfx1250) — compile-verified
//
#include <hip/hip_runtime.h>

#define B_  2
#define S_  2048
#define D_  1024
#define H_  16
#define DH_ 64

typedef float v2f __attribute__((ext_vector_type(2)));
typedef float v8f __attribute__((ext_vector_type(8)));

// fp32 WMMA: D(16x16,f32) = A(16x4,f32) x B(4x16,f32) + C
// A/B operand: 2 VGPRs per lane. Lanes 0-15: K = {0,1}; lanes 16-31: K = {2,3}.
// C/D layout: lanes 0-15 -> rows 0-7 (vgpr index = row), lanes 16-31 -> rows 8-15; col = lane%16.
__device__ __forceinline__ v8f wmma_f32(v2f a, v2f b, v8f c) {
  return __builtin_amdgcn_wmma_f32_16x16x4_f32(
      /*neg_a=*/false, a, /*neg_b=*/false, b,
      /*c_mod=*/(short)0, c, /*reuse_a=*/false, /*reuse_b=*/false);
}

// ---------------------------------------------------------------------------
// Stage 1: fused QKV projections, 64x64 register-blocked tile per wave.
// ws layout: [3][B][H][S][DH]  (per-head matrices contiguous, row-major S x 64)
// Per K-step of 4: 4 x b64 A-loads + 8 scalar B-loads -> 16 fp32 WMMAs.
// A 64-wide column tile spans exactly one head -> trivial epilogue scatter.
// ---------------------------------------------------------------------------
__global__ __launch_bounds__(256) void proj_kernel(
    const float* __restrict__ q, const float* __restrict__ k,
    const float* __restrict__ v,
    const float* __restrict__ Wq, const float* __restrict__ bq,
    const float* __restrict__ Wk, const float* __restrict__ bk,
    const float* __restrict__ Wv, const float* __restrict__ bv,
    float* __restrict__ ws) {
  const int lane   = threadIdx.x & 31;
  const int waveId = threadIdx.x >> 5;
  const int ROWT = (B_ * S_) / 64;   // 64 row tiles
  const int COLT = D_ / 64;          // 16 col tiles
  const int TPM  = ROWT * COLT;      // 1024 tiles per matrix

  int tile = blockIdx.x * 8 + waveId;          // 0 .. 3*TPM-1 (exact)
  int mat  = tile / TPM;
  int t    = tile % TPM;
  int rt   = t / COLT;
  int ct   = t % COLT;

  const float *X, *W, *bias;
  if (mat == 0)      { X = q; W = Wq; bias = bq; }
  else if (mat == 1) { X = k; W = Wk; bias = bk; }
  else               { X = v; W = Wv; bias = bv; }
  float* O = ws + (size_t)mat * (B_ * S_ * D_);

  const int ln     = lane & 15;
  const int koff   = (lane >= 16) ? 2 : 0;
  const int rowadd = (lane >= 16) ? 8 : 0;
  const int row0   = rt * 64;
  const int col0   = ct * 64;

  v8f acc[4][4];
#pragma unroll
  for (int i = 0; i < 4; ++i)
#pragma unroll
    for (int j = 0; j < 4; ++j) acc[i][j] = (v8f){};

  for (int k0 = 0; k0 < D_; k0 += 4) {
    v2f a[4];
#pragma unroll
    for (int i = 0; i < 4; ++i)
      a[i] = *(const v2f*)(X + (size_t)(row0 + i * 16 + ln) * D_ + k0 + koff);
    v2f bb[4];
#pragma unroll
    for (int j = 0; j < 4; ++j) {
      const int col = col0 + j * 16 + ln;
      bb[j].x = W[(size_t)(k0 + koff)     * D_ + col];
      bb[j].y = W[(size_t)(k0 + koff + 1) * D_ + col];
    }
#pragma unroll
    for (int i = 0; i < 4; ++i)
#pragma unroll
      for (int j = 0; j < 4; ++j)
        acc[i][j] = wmma_f32(a[i], bb[j], acc[i][j]);
  }

  // Epilogue: whole tile lives in one (batch, head) slab.
  const int h  = col0 / DH_;        // 64-wide tile == one head
  const int bi = row0 / S_;         // 64-row tile never crosses batch
  const int s0 = row0 % S_;
  float* Ohead = O + ((size_t)bi * H_ + h) * S_ * DH_;
  float bj[4];
#pragma unroll
  for (int j = 0; j < 4; ++j) bj[j] = bias[col0 + j * 16 + ln];
#pragma unroll
  for (int i = 0; i < 4; ++i) {
#pragma unroll
    for (int r = 0; r < 8; ++r) {
      const int s = s0 + i * 16 + r + rowadd;
#pragma unroll
      for (int j = 0; j < 4; ++j)
        Ohead[(size_t)s * DH_ + (j * 16 + ln)] = acc[i][j][r] + bj[j];
    }
  }
}

// ---------------------------------------------------------------------------
// Stage 2: flash attention. One wave owns a 32-query tile of one (b,h):
// two 16-row groups share every K-tile and V-tile B-operand (2x WMMA per load).
// Online softmax; 2 x 16x64 fp32 accumulators in registers; P re-layout via LDS.
// ---------------------------------------------------------------------------
__global__ __launch_bounds__(128) void attn_kernel(
    const float* __restrict__ ws, const int* __restrict__ mask,
    float* __restrict__ out) {
  __shared__ float lds[4][32][17];   // per-wave 32x16 P tile (padded)

  const int lane = threadIdx.x & 31;
  const int wv   = threadIdx.x >> 5;
  const int QT   = S_ / 32;          // 64 query tiles per head

  int tile = blockIdx.x * 4 + wv;    // 0..2047 (exact)
  int b    = tile / (H_ * QT);
  int rem  = tile % (H_ * QT);
  int h    = rem / QT;
  int qt   = rem % QT;

  const size_t headElems = (size_t)B_ * H_ * S_ * DH_;
  const float* Qh = ws + (size_t)(b * H_ + h) * S_ * DH_;
  const float* Kh = Qh + headElems;
  const float* Vh = Kh + headElems;
  const int*   mrow = mask + b * S_;

  const int ln     = lane & 15;
  const int koff   = (lane >= 16) ? 2 : 0;
  const int rowadd = (lane >= 16) ? 8 : 0;

  // Cache both 16x64 Q row-groups' A-operands (64 VGPRs).
  v2f qreg[2][16];
#pragma unroll
  for (int g = 0; g < 2; ++g) {
    const float* qrow = Qh + (size_t)(qt * 32 + g * 16 + ln) * DH_;
#pragma unroll
    for (int ks = 0; ks < 16; ++ks)
      qreg[g][ks] = *(const v2f*)(qrow + ks * 4 + koff);
  }

  v8f acc[2][4];
#pragma unroll
  for (int g = 0; g < 2; ++g)
#pragma unroll
    for (int t = 0; t < 4; ++t) acc[g][t] = (v8f){};
  float mrun[2][8], lrun[2][8];
#pragma unroll
  for (int g = 0; g < 2; ++g)
#pragma unroll
    for (int r = 0; r < 8; ++r) { mrun[g][r] = -1e30f; lrun[g][r] = 0.0f; }

  const float inv_s10 = 0.31622776601683794f;   // 1/sqrt(10)

  for (int kt = 0; kt < S_ / 16; ++kt) {
    const int key = kt * 16 + ln;               // this lane's key column

    // ---- scores: two 16x16 tiles of Q.K^T, K-tile operand shared ----
    v8f c[2];
    c[0] = (v8f){}; c[1] = (v8f){};
    const float* krow = Kh + (size_t)key * DH_;
#pragma unroll
    for (int ks = 0; ks < 16; ++ks) {
      v2f kv = *(const v2f*)(krow + ks * 4 + koff);   // B[k][n] = K[n][k]
      c[0] = wmma_f32(qreg[0][ks], kv, c[0]);
      c[1] = wmma_f32(qreg[1][ks], kv, c[1]);
    }

    // ---- scale + key mask (mask is per key position) ----
    const int mv = mrow[key];
#pragma unroll
    for (int g = 0; g < 2; ++g)
#pragma unroll
      for (int r = 0; r < 8; ++r)
        c[g][r] = mv ? c[g][r] * inv_s10 : -1e9f;

    // ---- online softmax (row reductions across the 16-lane half) ----
#pragma unroll
    for (int g = 0; g < 2; ++g) {
#pragma unroll
      for (int r = 0; r < 8; ++r) {
        float rm = c[g][r];
        rm = fmaxf(rm, __shfl_xor(rm, 8, 16));
        rm = fmaxf(rm, __shfl_xor(rm, 4, 16));
        rm = fmaxf(rm, __shfl_xor(rm, 2, 16));
        rm = fmaxf(rm, __shfl_xor(rm, 1, 16));
        float newm = fmaxf(mrun[g][r], rm);
        float p = __expf(c[g][r] - newm);
        float rs = p;
        rs += __shfl_xor(rs, 8, 16);
        rs += __shfl_xor(rs, 4, 16);
        rs += __shfl_xor(rs, 2, 16);
        rs += __shfl_xor(rs, 1, 16);
        float corr = __expf(mrun[g][r] - newm);
        lrun[g][r] = lrun[g][r] * corr + rs;
        mrun[g][r] = newm;
        c[g][r] = p;
#pragma unroll
        for (int t = 0; t < 4; ++t) acc[g][t][r] *= corr;
      }
    }

    // ---- bounce P through LDS: C-layout -> A-layout (both row groups) ----
#pragma unroll
    for (int g = 0; g < 2; ++g)
#pragma unroll
      for (int r = 0; r < 8; ++r)
        lds[wv][g * 16 + r + rowadd][ln] = c[g][r];
    // same-wave DS ops are in-order; compiler inserts s_wait_dscnt for the reads
    v2f pa[2][4];
#pragma unroll
    for (int g = 0; g < 2; ++g)
#pragma unroll
      for (int ks = 0; ks < 4; ++ks) {
        pa[g][ks].x = lds[wv][g * 16 + ln][ks * 4 + koff];
        pa[g][ks].y = lds[wv][g * 16 + ln][ks * 4 + koff + 1];
      }

    // ---- acc(32x64) += P(32x16) x V(16x64): V-tile operand shared ----
    const float* vbase = Vh + (size_t)(kt * 16) * DH_;
#pragma unroll
    for (int t = 0; t < 4; ++t) {
#pragma unroll
      for (int ks = 0; ks < 4; ++ks) {
        const int kk = ks * 4 + koff;
        v2f bvec;
        bvec.x = vbase[(size_t)kk       * DH_ + t * 16 + ln];
        bvec.y = vbase[(size_t)(kk + 1) * DH_ + t * 16 + ln];
        acc[0][t] = wmma_f32(pa[0][ks], bvec, acc[0][t]);
        acc[1][t] = wmma_f32(pa[1][ks], bvec, acc[1][t]);
      }
    }
  }

  // ---- epilogue: normalize and scatter to [B,S,D] ----
#pragma unroll
  for (int g = 0; g < 2; ++g) {
#pragma unroll
    for (int r = 0; r < 8; ++r) {
      const float invl = 1.0f / lrun[g][r];
      const int s = qt * 32 + g * 16 + r + rowadd;
#pragma unroll
      for (int t = 0; t < 4; ++t) {
        const int dcol = t * 16 + ln;
        out[((size_t)(b * S_ + s)) * D_ + h * DH_ + dcol] = acc[g][t][r] * invl;
      }
    }
  }
}

// ---------------------------------------------------------------------------
extern "C" void kernel_launch(void* const* d_in, const int* in_sizes, int n_in,
                              void* d_out, int out_size, void* d_ws, size_t ws_size,
                              hipStream_t stream) {
  (void)in_sizes; (void)n_in; (void)out_size; (void)ws_size;
  const float* q    = (const float*)d_in[0];
  const float* k    = (const float*)d_in[1];
  const float* v    = (const float*)d_in[2];
  const int*   mask = (const int*)d_in[3];
  const float* Wq   = (const float*)d_in[4];
  const float* bq   = (const float*)d_in[5];
  const float* Wk   = (const float*)d_in[6];
  const float* bk   = (const float*)d_in[7];
  const float* Wv   = (const float*)d_in[8];
  const float* bv   = (const float*)d_in[9];
  float* ws  = (float*)d_ws;   // needs 3*B*S*D*4 = 48 MiB
  float* out = (float*)d_out;

  // 3 matrices * 64 row-tiles * 16 col-tiles = 3072 waves, 8 waves/block
  proj_kernel<<<384, 256, 0, stream>>>(q, k, v, Wq, bq, Wk, bk, Wv, bv, ws);
  // B*H*(S/32) = 2048 query-tile waves, 4 waves/block
  attn_kernel<<<512, 128, 0, stream>>>(ws, mask, out);
}